// GINNet_20804821581835
// MI455X (gfx1250) — compile-verified
//
#include <hip/hip_runtime.h>
#include <hip/hip_bf16.h>

typedef __attribute__((ext_vector_type(2))) float v2f;
typedef __attribute__((ext_vector_type(8))) float v8f;

#define GIN_HC 64   // hidden channels == num classes == 64

// ---------------------------------------------------------------------------
// zero fill
// ---------------------------------------------------------------------------
__global__ void zero_f32(float* __restrict__ p, long long n) {
  long long i = (long long)blockIdx.x * blockDim.x + threadIdx.x;
  if (i < n) p[i] = 0.0f;
}

// ---------------------------------------------------------------------------
// scatter-add aggregation: agg[dst[e]][c] += h[src[e]][c]
// one thread = one (edge, 4-feature chunk); edge_index is int64 [2, E] flat.
// ---------------------------------------------------------------------------
template <int F>
__global__ void scatter_add_f(const long long* __restrict__ ei,
                              const float* __restrict__ h,
                              float* __restrict__ agg, int E) {
  const int CH = F / 4;
  long long gid = (long long)blockIdx.x * blockDim.x + threadIdx.x;
  long long total = (long long)E * CH;
  if (gid >= total) return;
  int e = (int)(gid / CH);
  int c = (int)(gid % CH) * 4;
  int s = (int)ei[e];
  int d = (int)ei[(long long)E + e];
  const float* hs = h + (long long)s * F + c;
  float* ad = agg + (long long)d * F + c;
  atomicAdd(ad + 0, hs[0]);
  atomicAdd(ad + 1, hs[1]);
  atomicAdd(ad + 2, hs[2]);
  atomicAdd(ad + 3, hs[3]);
}

// ---------------------------------------------------------------------------
// fp32 WMMA 16x16 tile accumulate over K (step 4): D += A(16xK) * B(Kx16)
// A from LDS (row-major, leading dim lda), B from global (row-major, ld 64).
// Per ISA layout: A lanes 0-15 hold K={k,k+1}, lanes 16-31 hold K={k+2,k+3};
// B symmetric with N striped across lanes.
// ---------------------------------------------------------------------------
template <int K>
__device__ __forceinline__ v8f wmma_f32_tile(const float* __restrict__ A, int lda,
                                             const float* __restrict__ B, int ldb,
                                             int lane, v8f acc) {
  const int row = lane & 15;
  const int col = lane & 15;
  const int kk = (lane >> 4) << 1;
#pragma unroll
  for (int k = 0; k < K; k += 4) {
    v2f a, b;
    a.x = A[row * lda + k + kk];
    a.y = A[row * lda + k + kk + 1];
    b.x = B[(k + kk) * ldb + col];
    b.y = B[(k + kk + 1) * ldb + col];
    acc = __builtin_amdgcn_wmma_f32_16x16x4_f32(
        /*neg_a=*/false, a, /*neg_b=*/false, b,
        /*c_mod=*/(short)0, acc, /*reuse_a=*/false, /*reuse_b=*/false);
  }
  return acc;
}

// ---------------------------------------------------------------------------
// Fused GIN block: Z = (1+eps)*x + agg; H = relu(Z@Wa + ba); OUT = H@Wb + bb
// One wave handles 16 nodes. 4 waves / block, disjoint per-wave LDS, no
// block-level barriers (only wave-local s_wait_dscnt fences), so wave-uniform
// early-exit is safe and EXEC is all-ones at every WMMA.
// ---------------------------------------------------------------------------
template <int NFIN, bool RELU_OUT>
__global__ void __launch_bounds__(128)
gin_mlp(const float* __restrict__ xin, const float* __restrict__ agg,
        const float* __restrict__ eps,
        const float* __restrict__ Wa, const float* __restrict__ ba,
        const float* __restrict__ Wb, const float* __restrict__ bb,
        float* __restrict__ out1, float* __restrict__ out2, int n) {
  const int NW = 4;
  __shared__ float Zs[NW][16 * (NFIN + 1)];   // padded stride: avoid bank conflicts
  __shared__ float Hs[NW][16 * (GIN_HC + 1)];

  const int lane = threadIdx.x & 31;
  const int w = threadIdx.x >> 5;
  const int tile = blockIdx.x * NW + w;
  const int ntiles = (n + 15) >> 4;
  if (tile >= ntiles) return;  // wave-uniform
  const int node0 = tile * 16;

  float* Z = Zs[w];
  float* H = Hs[w];
  const float ep = 1.0f + eps[0];

  // ---- stage Z = (1+eps)*x + agg into LDS (16 x NFIN) ----
  for (int idx = lane; idx < 16 * NFIN; idx += 32) {
    int r = idx / NFIN, c = idx - r * NFIN;
    int node = node0 + r;
    float v = 0.0f;
    if (node < n) {
      long long o = (long long)node * NFIN + c;
      v = ep * xin[o] + agg[o];
    }
    Z[r * (NFIN + 1) + c] = v;
  }
  asm volatile("s_wait_dscnt 0" ::: "memory");

  const int col = lane & 15;
  const int mhi = (lane >> 4) << 3;  // D layout: M = vgpr + 8*(lane>=16)

  // ---- H = relu(Z @ Wa + ba), 16 x 64 ----
#pragma unroll
  for (int nt = 0; nt < 4; ++nt) {
    v8f acc = {};
    acc = wmma_f32_tile<NFIN>(Z, NFIN + 1, Wa + nt * 16, GIN_HC, lane, acc);
    float bias = ba[nt * 16 + col];
#pragma unroll
    for (int i = 0; i < 8; ++i) {
      float v = acc[i] + bias;
      H[(mhi + i) * (GIN_HC + 1) + nt * 16 + col] = v > 0.0f ? v : 0.0f;
    }
  }
  asm volatile("s_wait_dscnt 0" ::: "memory");

  // ---- OUT = H @ Wb + bb, 16 x 64 ----
#pragma unroll
  for (int nt = 0; nt < 4; ++nt) {
    v8f acc = {};
    acc = wmma_f32_tile<GIN_HC>(H, GIN_HC + 1, Wb + nt * 16, GIN_HC, lane, acc);
    float bias = bb[nt * 16 + col];
#pragma unroll
    for (int i = 0; i < 8; ++i) {
      int node = node0 + mhi + i;
      if (node < n) {
        float v = acc[i] + bias;
        long long o = (long long)node * GIN_HC + nt * 16 + col;
        out1[o] = v;
        if (RELU_OUT) out2[o] = v > 0.0f ? v : 0.0f;
      }
    }
  }
}

// ---------------------------------------------------------------------------
// launch
// ---------------------------------------------------------------------------
extern "C" void kernel_launch(void* const* d_in, const int* in_sizes, int n_in,
                              void* d_out, int out_size, void* d_ws, size_t ws_size,
                              hipStream_t stream) {
  const float* x = (const float*)d_in[0];
  const long long* ei = (const long long*)d_in[1];  // int64 [2, E]
  const float* eps1 = (const float*)d_in[2];
  const float* eps2 = (const float*)d_in[3];
  const float* W1a = (const float*)d_in[4];
  const float* b1a = (const float*)d_in[5];
  const float* W1b = (const float*)d_in[6];
  const float* b1b = (const float*)d_in[7];
  const float* W2a = (const float*)d_in[8];
  const float* b2a = (const float*)d_in[9];
  const float* W2b = (const float*)d_in[10];
  const float* b2b = (const float*)d_in[11];

  const int NF = 32;
  const int n = in_sizes[0] / NF;
  const int E = in_sizes[1] / 2;

  // output: (logits [n,64], emb [n,64]) concatenated flat
  float* logits = (float*)d_out;
  float* emb = (float*)d_out + (long long)n * GIN_HC;

  // workspace: agg1 [n,32] | hbuf [n,64] | agg2 [n,64]
  float* agg1 = (float*)d_ws;
  float* hbuf = agg1 + (long long)n * NF;
  float* agg2 = hbuf + (long long)n * GIN_HC;

  // zero aggregation buffers
  {
    long long n1 = (long long)n * NF;
    long long n2 = (long long)n * GIN_HC;
    zero_f32<<<(int)((n1 + 255) / 256), 256, 0, stream>>>(agg1, n1);
    zero_f32<<<(int)((n2 + 255) / 256), 256, 0, stream>>>(agg2, n2);
  }

  // layer 1: agg1[dst] += x[src]
  {
    long long total = (long long)E * (NF / 4);
    scatter_add_f<32><<<(int)((total + 255) / 256), 256, 0, stream>>>(ei, x, agg1, E);
  }

  // layer 1 MLP: emb = MLP1((1+eps1)x + agg1); hbuf = relu(emb)
  {
    int ntiles = (n + 15) / 16;
    int blocks = (ntiles + 3) / 4;
    gin_mlp<32, true><<<blocks, 128, 0, stream>>>(x, agg1, eps1, W1a, b1a, W1b,
                                                  b1b, emb, hbuf, n);
  }

  // layer 2: agg2[dst] += hbuf[src]
  {
    long long total = (long long)E * (GIN_HC / 4);
    scatter_add_f<64><<<(int)((total + 255) / 256), 256, 0, stream>>>(ei, hbuf, agg2, E);
  }

  // layer 2 MLP: logits = MLP2((1+eps2)hbuf + agg2)
  {
    int ntiles = (n + 15) / 16;
    int blocks = (ntiles + 3) / 4;
    gin_mlp<64, false><<<blocks, 128, 0, stream>>>(hbuf, agg2, eps2, W2a, b2a,
                                                   W2b, b2b, logits, nullptr, n);
  }
}